// GATEncoder_28458453303310
// MI455X (gfx1250) — compile-verified
//
#include <hip/hip_runtime.h>
#include <math.h>

typedef __attribute__((ext_vector_type(2))) float v2f;
typedef __attribute__((ext_vector_type(8))) float v8f;

// ---------------------------------------------------------------------------
// D[nRows x C] = A[nRows x 128] @ B[128 x C], all row-major fp32.
// One wave32 computes one 16x16 tile of D with V_WMMA_F32_16X16X4_F32,
// chaining K=128 as 32 k-steps of 4.
//   A frag (16x4):  a[i] = A[lane&15][kh+i],        kh = (lane>>4)*2
//   B frag (4x16):  b[i] = B[kh+i][lane&15]
//   C/D  (16x16):   acc[j] = D[j + (lane>>4)*8][lane&15]
// ---------------------------------------------------------------------------
template <int C>
__global__ __launch_bounds__(128) void wmma_gemm_k128(
    const float* __restrict__ A, const float* __restrict__ B,
    float* __restrict__ D, int nRows)
{
  const int lane    = threadIdx.x & 31;
  const int wave    = threadIdx.x >> 5;
  const int rowTile = blockIdx.x;
  const int colTile = blockIdx.y * 4 + wave;   // wave-uniform, always < C/16

  const int m  = lane & 15;
  const int kh = (lane >> 4) << 1;             // 0 or 2

  int arow = rowTile * 16 + m;
  if (arow >= nRows) arow = nRows - 1;         // clamp tail (branchless select)
  const float* __restrict__ Ap = A + (size_t)arow * 128;
  const float* __restrict__ Bp = B + (size_t)colTile * 16 + m;

  v8f acc = {};
#pragma unroll
  for (int k = 0; k < 128; k += 4) {
    v2f a, b;
    a.x = Ap[k + kh];
    a.y = Ap[k + kh + 1];
    b.x = Bp[(size_t)(k + kh) * C];
    b.y = Bp[(size_t)(k + kh + 1) * C];
    // 8 args: (neg_a, A, neg_b, B, c_mod, C, reuse_a, reuse_b)
    acc = __builtin_amdgcn_wmma_f32_16x16x4_f32(false, a, false, b,
                                                (short)0, acc, false, false);
  }

  const int rbase = rowTile * 16 + ((lane >> 4) << 3);
  float* __restrict__ Dp = D + (size_t)rbase * C + (size_t)colTile * 16 + m;
#pragma unroll
  for (int j = 0; j < 8; ++j)
    if (rbase + j < nRows) Dp[(size_t)j * C] = acc[j];
}

// ---------------------------------------------------------------------------
// Per-node attention logits: als[n] = <h[n,:], a_src>, ald[n] = <h[n,:], a_dst>
// One wave32 per node, shuffle tree reduction.
// ---------------------------------------------------------------------------
template <int C>
__global__ void attn_logits(const float* __restrict__ h,
                            const float* __restrict__ asrc,
                            const float* __restrict__ adst,
                            float* __restrict__ als, float* __restrict__ ald,
                            int nN)
{
  const int node = (int)((blockIdx.x * (size_t)blockDim.x + threadIdx.x) >> 5);
  const int lane = threadIdx.x & 31;
  if (node >= nN) return;
  const float* __restrict__ row = h + (size_t)node * C;
  float ss = 0.0f, sd = 0.0f;
#pragma unroll
  for (int c = lane; c < C; c += 32) {
    float v = row[c];
    ss += v * asrc[c];
    sd += v * adst[c];
  }
#pragma unroll
  for (int off = 16; off > 0; off >>= 1) {
    ss += __shfl_down(ss, off, 32);
    sd += __shfl_down(sd, off, 32);
  }
  if (lane == 0) { als[node] = ss; ald[node] = sd; }
}

// ---------------------------------------------------------------------------
// Utility inits (d_out / d_ws are poisoned, must re-init every call)
// ---------------------------------------------------------------------------
__global__ void fill_f32(float* __restrict__ p, float v, int n) {
  int i = blockIdx.x * blockDim.x + threadIdx.x;
  if (i < n) p[i] = v;
}
__global__ void init_rows(float* __restrict__ p, const float* __restrict__ b,
                          int n, int mask) {
  int i = blockIdx.x * blockDim.x + threadIdx.x;
  if (i < n) p[i] = b[i & mask];
}
__global__ void bias_relu(float* __restrict__ p, const float* __restrict__ b,
                          int n, int mask) {
  int i = blockIdx.x * blockDim.x + threadIdx.x;
  if (i < n) p[i] = fmaxf(p[i] + b[i & mask], 0.0f);
}

// ---------------------------------------------------------------------------
// Edge kernels. Edge ids [0,nE) come from edge_index; ids [nE, nE+nN) are
// implicit self-loops (src = dst = id - nE).
// ---------------------------------------------------------------------------
__global__ void edge_score(const int* __restrict__ ei,
                           const float* __restrict__ als,
                           const float* __restrict__ ald,
                           float* __restrict__ esc, float* __restrict__ mmax,
                           int nE, int nN)
{
  int e = blockIdx.x * blockDim.x + threadIdx.x;
  if (e >= nE + nN) return;
  int s, d;
  if (e < nE) { s = ei[e]; d = ei[nE + e]; } else { s = e - nE; d = s; }
  float v = als[s] + ald[d];
  v = (v > 0.0f) ? v : 0.2f * v;          // LeakyReLU(0.2)
  esc[e] = v;
  atomicMax(mmax + d, v);                  // segment max
}

__global__ void edge_exp(const int* __restrict__ ei,
                         const float* __restrict__ mmax,
                         float* __restrict__ esc, float* __restrict__ denom,
                         int nE, int nN)
{
  int e = blockIdx.x * blockDim.x + threadIdx.x;
  if (e >= nE + nN) return;
  int d = (e < nE) ? ei[nE + e] : (e - nE);
  float ex = __expf(esc[e] - mmax[d]);
  esc[e] = ex;
  atomicAdd(denom + d, ex);                // segment sum
}

// One wave32 per edge: 512B coalesced gather of h[src], fp32 atomic scatter.
template <int C>
__global__ void edge_aggregate(const int* __restrict__ ei,
                               const float* __restrict__ esc,
                               const float* __restrict__ denom,
                               const float* __restrict__ h,
                               float* __restrict__ out,
                               int nE, int nN)
{
  const int e    = (int)((blockIdx.x * (size_t)blockDim.x + threadIdx.x) >> 5);
  const int lane = threadIdx.x & 31;
  if (e >= nE + nN) return;
  int s, d;
  if (e < nE) { s = ei[e]; d = ei[nE + e]; } else { s = e - nE; d = s; }
  const float alpha = esc[e] / (denom[d] + 1e-16f);
  const float* __restrict__ hs = h + (size_t)s * C;
  float* __restrict__ od = out + (size_t)d * C;
  if constexpr (C == 128) {
    float4 v = reinterpret_cast<const float4*>(hs)[lane];
    atomicAdd(od + 4 * lane + 0, alpha * v.x);
    atomicAdd(od + 4 * lane + 1, alpha * v.y);
    atomicAdd(od + 4 * lane + 2, alpha * v.z);
    atomicAdd(od + 4 * lane + 3, alpha * v.w);
  } else {
    float2 v = reinterpret_cast<const float2*>(hs)[lane];
    atomicAdd(od + 2 * lane + 0, alpha * v.x);
    atomicAdd(od + 2 * lane + 1, alpha * v.y);
  }
}

// ---------------------------------------------------------------------------
extern "C" void kernel_launch(void* const* d_in, const int* in_sizes, int n_in,
                              void* d_out, int out_size, void* d_ws, size_t ws_size,
                              hipStream_t stream)
{
  const float* x   = (const float*)d_in[0];
  const int*   ei  = (const int*)  d_in[1];
  const float* W1  = (const float*)d_in[2];
  const float* as1 = (const float*)d_in[3];
  const float* ad1 = (const float*)d_in[4];
  const float* b1  = (const float*)d_in[5];
  const float* W2  = (const float*)d_in[6];
  const float* as2 = (const float*)d_in[7];
  const float* ad2 = (const float*)d_in[8];
  const float* b2  = (const float*)d_in[9];
  float* out = (float*)d_out;

  const int nN   = in_sizes[0] / 128;   // 50000
  const int nE   = in_sizes[1] / 2;     // 800000
  const int nTot = nE + nN;             // + self loops

  // workspace layout (fp32):  h1[nN*128] | z[nN*128] | als | ald | mmax | denom | esc
  float* ws    = (float*)d_ws;
  float* h1    = ws;
  float* z     = h1 + (size_t)nN * 128;
  float* als   = z  + (size_t)nN * 128;
  float* ald   = als + nN;
  float* mmax  = ald + nN;
  float* denom = mmax + nN;
  float* esc   = denom + nN;
  float* h2    = h1;                    // h1 dead after layer-1 aggregation

  const int rowTiles = (nN + 15) / 16;
  const int TB = 256;
  const int gN    = (nN + TB - 1) / TB;
  const int gN128 = (nN * 128 + TB - 1) / TB;
  const int gN64  = (nN * 64 + TB - 1) / TB;
  const int gEdge = (nTot + TB - 1) / TB;
  const int gWaveN = (nN + 7) / 8;       // 8 waves / 256-thread block
  const int gWaveE = (nTot + 7) / 8;

  // ----- layer 1 : h1 = x @ W1 (128->128) -----
  wmma_gemm_k128<128><<<dim3(rowTiles, 2), 128, 0, stream>>>(x, W1, h1, nN);
  attn_logits<128><<<gWaveN, TB, 0, stream>>>(h1, as1, ad1, als, ald, nN);
  fill_f32<<<gN, TB, 0, stream>>>(mmax, -INFINITY, nN);
  fill_f32<<<gN, TB, 0, stream>>>(denom, 0.0f, nN);
  fill_f32<<<gN128, TB, 0, stream>>>(z, 0.0f, nN * 128);
  edge_score<<<gEdge, TB, 0, stream>>>(ei, als, ald, esc, mmax, nE, nN);
  edge_exp  <<<gEdge, TB, 0, stream>>>(ei, mmax, esc, denom, nE, nN);
  edge_aggregate<128><<<gWaveE, TB, 0, stream>>>(ei, esc, denom, h1, z, nE, nN);
  bias_relu<<<gN128, TB, 0, stream>>>(z, b1, nN * 128, 127);

  // ----- layer 2 : h2 = z @ W2 (128->64) -----
  wmma_gemm_k128<64><<<dim3(rowTiles, 1), 128, 0, stream>>>(z, W2, h2, nN);
  attn_logits<64><<<gWaveN, TB, 0, stream>>>(h2, as2, ad2, als, ald, nN);
  fill_f32<<<gN, TB, 0, stream>>>(mmax, -INFINITY, nN);
  fill_f32<<<gN, TB, 0, stream>>>(denom, 0.0f, nN);
  init_rows<<<gN64, TB, 0, stream>>>(out, b2, nN * 64, 63);   // out = b2 (broadcast)
  edge_score<<<gEdge, TB, 0, stream>>>(ei, als, ald, esc, mmax, nE, nN);
  edge_exp  <<<gEdge, TB, 0, stream>>>(ei, mmax, esc, denom, nE, nN);
  edge_aggregate<64><<<gWaveE, TB, 0, stream>>>(ei, esc, denom, h2, out, nE, nN);
}